// TokenMask_79362405695963
// MI455X (gfx1250) — compile-verified
//
#include <hip/hip_runtime.h>
#include <hip/hip_bf16.h>
#include <stdint.h>

// Problem constants (from reference)
#define BB 64
#define TT 2048
#define DD 1024
#define MASK_RATIO 0.75f

// ---------------------------------------------------------------------------
// Kernel 1: per-(b) row -> keep mask, argmax fix, stats.  One 256-thread block
// per batch row; each thread scans 8 tokens, then LDS tree reduction.
// Writes keep_mask (float 0/1) and the 3 stats vectors directly into d_out.
// ---------------------------------------------------------------------------
__global__ __launch_bounds__(256)
void TokenMask_keep_kernel(const int*   __restrict__ valid,
                           const float* __restrict__ rand_vals,
                           const float* __restrict__ scores,
                           float* __restrict__ keep_out,   // [B*T] floats
                           float* __restrict__ stats_out)  // [3*B] floats
{
    const int b   = blockIdx.x;
    const int tid = threadIdx.x;
    const int base = b * TT;

    int   vcnt = 0;          // valid count (this thread)
    int   kcnt = 0;          // kept count before fix (this thread)
    float best = -2.0f;      // masked choice score max (sentinel below -1.0)
    int   bidx = 0;

    for (int t = tid; t < TT; t += 256) {
        const int   v  = valid[base + t];
        const float r  = rand_vals[base + t];
        const float sc = (v != 0) ? scores[base + t] : -1.0f;  // matches jnp.where(valid, s, -1)
        const int   k  = (v != 0) && (r >= MASK_RATIO);
        vcnt += (v != 0);
        kcnt += k;
        keep_out[base + t] = k ? 1.0f : 0.0f;
        if (sc > best) { best = sc; bidx = t; }   // ascending t => first occurrence on ties
    }

    __shared__ float s_best[256];
    __shared__ int   s_bidx[256];
    __shared__ int   s_v[256];
    __shared__ int   s_k[256];
    s_best[tid] = best; s_bidx[tid] = bidx; s_v[tid] = vcnt; s_k[tid] = kcnt;
    __syncthreads();

    for (int off = 128; off > 0; off >>= 1) {
        if (tid < off) {
            s_v[tid] += s_v[tid + off];
            s_k[tid] += s_k[tid + off];
            const float ob = s_best[tid + off];
            const int   oi = s_bidx[tid + off];
            if (ob > s_best[tid] || (ob == s_best[tid] && oi < s_bidx[tid])) {
                s_best[tid] = ob; s_bidx[tid] = oi;
            }
        }
        __syncthreads();
    }

    if (tid == 0) {
        const int  vc        = s_v[0];
        int        kc        = s_k[0];
        const bool has_valid = (vc > 0);
        const bool needs_fix = has_valid && (kc == 0);
        if (needs_fix) {
            keep_out[base + s_bidx[0]] = 1.0f;   // guarantee one kept token
            kc = 1;
        }
        const float vcf   = (float)vc;
        const float kcf   = (float)kc;
        const float denom = vcf > 1.0f ? vcf : 1.0f;
        stats_out[0 * BB + b] = vcf;                 // valid_token_count
        stats_out[1 * BB + b] = kcf;                 // kept_token_count
        stats_out[2 * BB + b] = 1.0f - kcf / denom;  // actual_token_mask_ratio
    }
}

// ---------------------------------------------------------------------------
// Kernel 2: streaming apply.  One wave32 per 4KiB token row (8 rows / block).
// Kept row  : DMA-style copy through LDS with gfx1250 async ops
//             (global_load_async_to_lds_b128 / global_store_async_from_lds_b128,
//              tracked by ASYNCcnt, wave-local s_wait_asynccnt sync).
// Masked row: write zeros with global_store_b128 (no feature read -> ~75% of
//             the 512MiB feature traffic is skipped).
// Branch is wave-uniform (keep flag is per row), so EXEC stays all-ones for
// the async instructions as the ISA requires.
// ---------------------------------------------------------------------------
__global__ __launch_bounds__(256)
void TokenMask_apply_kernel(const float* __restrict__ feat,
                            const float* __restrict__ keepf,   // [B*T] 0.0/1.0
                            float* __restrict__ out)           // [B*T*D]
{
    const unsigned wave = threadIdx.x >> 5;
    const unsigned lane = threadIdx.x & 31u;
    const size_t   row  = (size_t)blockIdx.x * 8u + wave;

    const float k = keepf[row];

    __shared__ __align__(128) unsigned char smem[8 * 4096];   // 4KiB per wave

    const size_t row_bytes = row * (size_t)(DD * 4);
    if (k > 0.5f) {
        const uint32_t lbase = (uint32_t)(uintptr_t)(&smem[wave * 4096]) + lane * 16u;
        const unsigned long long gsrc =
            (unsigned long long)(uintptr_t)feat + row_bytes + (unsigned long long)lane * 16ull;
        const unsigned long long gdst =
            (unsigned long long)(uintptr_t)out + row_bytes + (unsigned long long)lane * 16ull;

        // 8 chunks x (32 lanes x 16B) = 4KiB row, all in flight on ASYNCcnt.
        #pragma unroll
        for (int c = 0; c < 8; ++c) {
            const uint32_t           la = lbase + (uint32_t)(c * 512);
            const unsigned long long ga = gsrc + (unsigned long long)(c * 512);
            asm volatile("global_load_async_to_lds_b128 %0, %1, off"
                         :: "v"(la), "v"(ga) : "memory");
        }
        asm volatile("s_wait_asynccnt 0x0" ::: "memory");   // loads landed in LDS
        #pragma unroll
        for (int c = 0; c < 8; ++c) {
            const uint32_t           la = lbase + (uint32_t)(c * 512);
            const unsigned long long ga = gdst + (unsigned long long)(c * 512);
            asm volatile("global_store_async_from_lds_b128 %0, %1, off"
                         :: "v"(ga), "v"(la) : "memory");
        }
        asm volatile("s_wait_asynccnt 0x0" ::: "memory");   // LDS safe to reuse / wave may end
    } else {
        float4* dst4 = (float4*)((char*)out + row_bytes);
        const float4 z = make_float4(0.f, 0.f, 0.f, 0.f);
        #pragma unroll
        for (int c = 0; c < 8; ++c) {
            dst4[lane + (unsigned)c * 32u] = z;             // global_store_b128
        }
    }
}

// ---------------------------------------------------------------------------
// Launch wrapper.
// Inputs (setup_inputs order): 0 token_features f32 [B,T,D], 1 valid_token_mask
// int [B,T], 2 rand_vals f32 [B,T], 3 choice_scores f32 [B,T].
// d_out layout (flat, return order): masked_tokens [B*T*D] | keep_mask [B*T] |
// valid_token_count [B] | kept_token_count [B] | actual_token_mask_ratio [B].
// ---------------------------------------------------------------------------
extern "C" void kernel_launch(void* const* d_in, const int* in_sizes, int n_in,
                              void* d_out, int out_size, void* d_ws, size_t ws_size,
                              hipStream_t stream) {
    (void)in_sizes; (void)n_in; (void)out_size; (void)d_ws; (void)ws_size;

    const float* feat   = (const float*)d_in[0];
    const int*   valid  = (const int*)  d_in[1];
    const float* rnd    = (const float*)d_in[2];
    const float* scores = (const float*)d_in[3];

    float* out        = (float*)d_out;
    float* keep_out   = out + (size_t)BB * TT * DD;   // [B*T]
    float* stats_out  = keep_out + (size_t)BB * TT;   // [3*B]

    // Pass 1: keep mask + fix + stats (tiny: 64 blocks)
    TokenMask_keep_kernel<<<BB, 256, 0, stream>>>(valid, rnd, scores, keep_out, stats_out);

    // Pass 2: stream rows; (B*T)/8 rows-per-block = 16384 blocks
    TokenMask_apply_kernel<<<(BB * TT) / 8, 256, 0, stream>>>(feat, keep_out, out);
}